// GCRNN_81174881894863
// MI455X (gfx1250) — compile-verified
//
#include <hip/hip_runtime.h>
#include <hip/hip_bf16.h>
#include <math.h>

#define U_NODES 100000
#define N_NODES 50000
#define E_EDGES 500000
#define DIM     128
#define GATES   512   // 4*DIM
#define MB      32    // users per LSTM block (2 WMMA M-tiles)
#define AROW    (DIM + 4)  // padded LDS row: 132 floats -> conflict-free frag reads

typedef float v2f __attribute__((ext_vector_type(2)));
typedef float v8f __attribute__((ext_vector_type(8)));

// ---------------------------------------------------------------------------
// Zero the accumulator / counter region of the workspace (must run per call:
// the harness only poisons ws once, and graphs replay this launch sequence).
// ---------------------------------------------------------------------------
__global__ __launch_bounds__(256) void zero_kernel(float* __restrict__ p, long n) {
  long i = (long)blockIdx.x * 256 + threadIdx.x;
  if (i < n) p[i] = 0.0f;
}

// ---------------------------------------------------------------------------
// One wave (32 lanes) per edge. Each lane owns a float4 slice of the 128-dim
// row: gather src row, multiply by edge row, float-atomic-add into acc[dst].
// Lane 0 also counts the edge for the mean. news/user tables are L2-resident
// (25.6/51.2 MB vs 192 MB L2) so the random gather and the atomics stay
// on-chip; only the edge features stream from HBM (2 x 256 MB).
// ---------------------------------------------------------------------------
__global__ __launch_bounds__(256) void scatter_kernel(
    const float* __restrict__ src_feat, const float* __restrict__ edge_feat,
    const int* __restrict__ src_idx, const int* __restrict__ dst_idx,
    float* __restrict__ acc, float* __restrict__ cnt, int nEdges) {
  const int wave = threadIdx.x >> 5;
  const int lane = threadIdx.x & 31;
  const int e = blockIdx.x * 8 + wave;
  if (e >= nEdges) return;
  const int s = src_idx[e];
  const int d = dst_idx[e];
  const float4 a = ((const float4*)(src_feat + (long)s * DIM))[lane];
  const float4 b = ((const float4*)(edge_feat + (long)e * DIM))[lane];
  float* out = acc + (long)d * DIM + lane * 4;
  atomicAdd(out + 0, a.x * b.x);
  atomicAdd(out + 1, a.y * b.y);
  atomicAdd(out + 2, a.z * b.z);
  atomicAdd(out + 3, a.w * b.w);
  if (lane == 0) atomicAdd(cnt + d, 1.0f);
}

// ---------------------------------------------------------------------------
// dst_new = dst_feat + (cnt > 0 ? acc/cnt : 0), float4-vectorized.
// ---------------------------------------------------------------------------
__global__ __launch_bounds__(256) void mean_update_kernel(
    const float* __restrict__ dst_feat, const float* __restrict__ acc,
    const float* __restrict__ cnt, float* __restrict__ out, int nNodes) {
  const long total = (long)nNodes * (DIM / 4);
  long i = (long)blockIdx.x * 256 + threadIdx.x;
  if (i >= total) return;
  const int node = (int)(i / (DIM / 4));
  const float c = cnt[node];
  const float inv = (c > 0.0f) ? (1.0f / c) : 0.0f;
  const float4 a = ((const float4*)acc)[i];
  const float4 f = ((const float4*)dst_feat)[i];
  float4 r;
  r.x = f.x + a.x * inv;
  r.y = f.y + a.y * inv;
  r.z = f.z + a.z * inv;
  r.w = f.w + a.w * inv;
  ((float4*)out)[i] = r;
}

__device__ __forceinline__ float sigf(float x) {
  return 1.0f / (1.0f + __expf(-x));
}

// ---------------------------------------------------------------------------
// LSTM cell via V_WMMA_F32_16X16X4_F32 (fp32 in/out, exact precision match).
// Block = 256 threads = 8 waves, handles 32 users (2 M-tiles).
// Each wave owns 64 of the 512 gate columns as 4 N-tiles -> 8 independent
// accumulator chains per wave (ILP to hide load latency).
// user_new / prev_hn tiles are staged in LDS once per block via coalesced
// b128 loads (8x less global A-traffic than per-wave strided reads); rows
// padded to 132 floats so ds_load_b64 fragment reads are bank-conflict-free.
// K loop (128, step 4): 4 LDS A-frags + 8 global B-frags feed 16 WMMAs/iter.
// A frag (16x4 f32, ISA layout): lanes 0-15 -> K = k,k+1 ; lanes 16-31 -> k+2,k+3.
// B frag (4x16): B[kk][n] = W[n][kk]; same half-wave K split, lane%16 -> N.
// C/D (16x16 f32): VGPR j -> M = j + 8*(lane>=16), N = lane%16.
// Gates staged in 64 KB LDS, then activations -> h, c.
// ---------------------------------------------------------------------------
__global__ __launch_bounds__(256) void lstm_wmma_kernel(
    const float* __restrict__ user_new,   // [U,128]
    const float* __restrict__ prev_hn,    // [U,128]
    const float* __restrict__ prev_cs,    // [U,128]
    const float* __restrict__ W_ih,       // [512,128] row-major
    const float* __restrict__ W_hh,       // [512,128] row-major
    const float* __restrict__ b_ih,       // [512]
    const float* __restrict__ b_hh,       // [512]
    float* __restrict__ h_out,            // [U,128]
    float* __restrict__ c_out) {          // [U,128]
  __shared__ float Atile[MB * AROW];      // 16.9 KB  user_new block
  __shared__ float Htile[MB * AROW];      // 16.9 KB  prev_hn block
  __shared__ float gates[MB * GATES];     // 64 KB    gate staging

  const int u0 = blockIdx.x * MB;         // U % 32 == 0 -> always full tile

  // --- Stage A/H tiles: 32 rows x 128 floats each, coalesced float4 copies
  for (int c = threadIdx.x; c < MB * (DIM / 4); c += 256) {
    const int r    = c >> 5;        // row 0..31   (DIM/4 == 32 chunks per row)
    const int col4 = c & 31;
    const float4 va = ((const float4*)(user_new + (long)(u0 + r) * DIM))[col4];
    *(float4*)&Atile[r * AROW + col4 * 4] = va;
    const float4 vh = ((const float4*)(prev_hn + (long)(u0 + r) * DIM))[col4];
    *(float4*)&Htile[r * AROW + col4 * 4] = vh;
  }
  __syncthreads();

  const int wave = threadIdx.x >> 5;
  const int lane = threadIdx.x & 31;
  const int half = lane >> 4;             // 0: lanes 0-15, 1: lanes 16-31
  const int l16  = lane & 15;
  const int nBase = wave * 64;

  // B-fragment base pointers for the 4 column tiles of each weight matrix
  const float* wI[4];
  const float* wH[4];
#pragma unroll
  for (int t = 0; t < 4; ++t) {
    const int n = nBase + t * 16 + l16;
    wI[t] = W_ih + (long)n * DIM + half * 2;
    wH[t] = W_hh + (long)n * DIM + half * 2;
  }

  // A-fragment LDS base pointers (per-lane row, per-half K offset)
  const float* a0p = &Atile[l16 * AROW + half * 2];        // users u0+0..15
  const float* a1p = &Atile[(16 + l16) * AROW + half * 2]; // users u0+16..31
  const float* h0p = &Htile[l16 * AROW + half * 2];
  const float* h1p = &Htile[(16 + l16) * AROW + half * 2];

  v8f acc[2][4];
#pragma unroll
  for (int m = 0; m < 2; ++m)
#pragma unroll
    for (int t = 0; t < 4; ++t) acc[m][t] = (v8f)(0.0f);

  for (int k = 0; k < DIM; k += 4) {
    const v2f a0 = *(const v2f*)(a0p + k);
    const v2f a1 = *(const v2f*)(a1p + k);
    const v2f h0 = *(const v2f*)(h0p + k);
    const v2f h1 = *(const v2f*)(h1p + k);
#pragma unroll
    for (int t = 0; t < 4; ++t) {
      const v2f bi = *(const v2f*)(wI[t] + k);
      const v2f bh = *(const v2f*)(wH[t] + k);
      acc[0][t] = __builtin_amdgcn_wmma_f32_16x16x4_f32(
          false, a0, false, bi, (short)0, acc[0][t], false, false);
      acc[1][t] = __builtin_amdgcn_wmma_f32_16x16x4_f32(
          false, a1, false, bi, (short)0, acc[1][t], false, false);
      acc[0][t] = __builtin_amdgcn_wmma_f32_16x16x4_f32(
          false, h0, false, bh, (short)0, acc[0][t], false, false);
      acc[1][t] = __builtin_amdgcn_wmma_f32_16x16x4_f32(
          false, h1, false, bh, (short)0, acc[1][t], false, false);
    }
  }

  // Spill accumulators (+ biases) to LDS in row-major [32][512]
#pragma unroll
  for (int t = 0; t < 4; ++t) {
    const int n = nBase + t * 16 + l16;
    const float bias = b_ih[n] + b_hh[n];
#pragma unroll
    for (int j = 0; j < 8; ++j) {
      const int m = j + half * 8;
      gates[m * GATES + n]        = acc[0][t][j] + bias;  // users u0+0..15
      gates[(16 + m) * GATES + n] = acc[1][t][j] + bias;  // users u0+16..31
    }
  }
  __syncthreads();

  // Activations: 32 users x 128 dims = 4096 elements, 16 per thread
  for (int idx = threadIdx.x; idx < MB * DIM; idx += 256) {
    const int m = idx >> 7;        // user row within tile
    const int d = idx & (DIM - 1); // feature dim
    const float gi = gates[m * GATES +   0 + d];
    const float gf = gates[m * GATES + 128 + d];
    const float gg = gates[m * GATES + 256 + d];
    const float go = gates[m * GATES + 384 + d];
    const long gidx = (long)(u0 + m) * DIM + d;
    const float cs = prev_cs[gidx];
    const float c = sigf(gf) * cs + sigf(gi) * tanhf(gg);
    const float h = sigf(go) * tanhf(c);
    h_out[gidx] = h;
    c_out[gidx] = c;
  }
}

// ---------------------------------------------------------------------------
// Launch sequence (all on `stream`, graph-capture safe: no mallocs/syncs).
// Workspace layout (floats):
//   [accU: U*D][cntU: U][accN: N*D][cntN: N][userNew: U*D]
// The first four regions are zeroed every call; userNew is fully overwritten.
// Output layout: [h: U*D][c: U*D][news_new: N*D]
// ---------------------------------------------------------------------------
extern "C" void kernel_launch(void* const* d_in, const int* in_sizes, int n_in,
                              void* d_out, int out_size, void* d_ws, size_t ws_size,
                              hipStream_t stream) {
  const float* user_feat = (const float*)d_in[0];
  const float* news_feat = (const float*)d_in[1];
  const float* edge_rev  = (const float*)d_in[2];
  const float* edge_fwd  = (const float*)d_in[3];
  const float* prev_hn   = (const float*)d_in[4];
  const float* prev_cs   = (const float*)d_in[5];
  const float* W_ih      = (const float*)d_in[6];
  const float* W_hh      = (const float*)d_in[7];
  const float* b_ih      = (const float*)d_in[8];
  const float* b_hh      = (const float*)d_in[9];
  const int*   user_idx  = (const int*)d_in[10];
  const int*   news_idx  = (const int*)d_in[11];

  float* ws      = (float*)d_ws;
  float* accU    = ws;
  float* cntU    = accU + (long)U_NODES * DIM;
  float* accN    = cntU + U_NODES;
  float* cntN    = accN + (long)N_NODES * DIM;
  float* userNew = cntN + N_NODES;

  float* out      = (float*)d_out;
  float* h_out    = out;
  float* c_out    = out + (long)U_NODES * DIM;
  float* news_out = c_out + (long)U_NODES * DIM;

  const long zN = (long)U_NODES * DIM + U_NODES + (long)N_NODES * DIM + N_NODES;
  zero_kernel<<<(int)((zN + 255) / 256), 256, 0, stream>>>(ws, zN);

  // reverse pass: news -> user
  scatter_kernel<<<E_EDGES / 8, 256, 0, stream>>>(
      news_feat, edge_rev, news_idx, user_idx, accU, cntU, E_EDGES);
  mean_update_kernel<<<(int)(((long)U_NODES * (DIM / 4) + 255) / 256), 256, 0, stream>>>(
      user_feat, accU, cntU, userNew, U_NODES);

  // forward pass: user -> news (uses updated user feats)
  scatter_kernel<<<E_EDGES / 8, 256, 0, stream>>>(
      userNew, edge_fwd, user_idx, news_idx, accN, cntN, E_EDGES);
  mean_update_kernel<<<(int)(((long)N_NODES * (DIM / 4) + 255) / 256), 256, 0, stream>>>(
      news_feat, accN, cntN, news_out, N_NODES);

  // LSTM cell on user nodes (WMMA f32, LDS-staged activations)
  lstm_wmma_kernel<<<U_NODES / MB, 256, 0, stream>>>(
      userNew, prev_hn, prev_cs, W_ih, W_hh, b_ih, b_hh, h_out, c_out);
}